// Memory_36979668418872
// MI455X (gfx1250) — compile-verified
//
#include <hip/hip_runtime.h>
#include <hip/hip_bf16.h>
#include <cstddef>

typedef __attribute__((ext_vector_type(2))) float v2f;
typedef __attribute__((ext_vector_type(8))) float v8f;

#define BS 64
#define NN 2048
#define DD 512

// ---------------------------------------------------------------------------
// Kernel A: content addressing partials.
//   dot[b,n]   = sum_d M[b,n,d] * k[b,d]        (via V_WMMA_F32_16X16X4_F32)
//   sumsq[b,n] = sum_d M[b,n,d]^2               (VALU + half-wave swap)
// One wave handles a 16-row tile; 4 waves per block. HBM-bound: 268 MB read.
// B operand is built branchlessly (v_cndmask) so the inner loop is pure
// load -> wmma with no EXEC save/restore. Two accumulators break the
// WMMA->WMMA RAW chain.
// ---------------------------------------------------------------------------
__global__ __launch_bounds__(128) void ntm_content_kernel(
    const float* __restrict__ M, const float* __restrict__ k,
    float* __restrict__ dot, float* __restrict__ sumsq)
{
    const int b    = blockIdx.y;
    const int wave = threadIdx.x >> 5;
    const int lane = threadIdx.x & 31;
    const int n0   = blockIdx.x * 64 + wave * 16;
    const int row  = n0 + (lane & 15);
    const int colSel = (lane >> 4) * 2;           // lanes 0-15: K=0,1 ; lanes 16-31: K=2,3
    const bool lead  = (lane & 15) == 0;          // B column 0 holders (lanes 0 and 16)

    const float* Mrow = M + ((size_t)b * NN + row) * DD + colSel;
    const float* kb   = k + (size_t)b * DD + colSel;

    v8f acc0 = {};         // C/D accumulators: 16x16 f32, we use column N=0
    v8f acc1 = {};
    float sq = 0.0f;

    #pragma unroll 8
    for (int d0 = 0; d0 < DD; d0 += 8) {
        // even k-step
        v2f a0 = *(const v2f*)(Mrow + d0);
        v2f k0 = *(const v2f*)(kb + d0);          // broadcast load, L0-cached
        v2f b0;
        b0.x = lead ? k0.x : 0.0f;                // v_cndmask, no branch
        b0.y = lead ? k0.y : 0.0f;
        // odd k-step
        v2f a1 = *(const v2f*)(Mrow + d0 + 4);
        v2f k1 = *(const v2f*)(kb + d0 + 4);
        v2f b1;
        b1.x = lead ? k1.x : 0.0f;
        b1.y = lead ? k1.y : 0.0f;

        acc0 = __builtin_amdgcn_wmma_f32_16x16x4_f32(
                   false, a0, false, b0, (short)0, acc0, false, false);
        acc1 = __builtin_amdgcn_wmma_f32_16x16x4_f32(
                   false, a1, false, b1, (short)0, acc1, false, false);

        sq += a0.x * a0.x + a0.y * a0.y;
        sq += a1.x * a1.x + a1.y * a1.y;
    }

    // sumsq: lane m has K={0,1} partials, lane m+16 has K={2,3} -> swap halves.
    sq += __shfl_xor(sq, 16, 32);
    if (lane < 16) sumsq[(size_t)b * NN + row] = sq;

    // dot: column N=0 of D. Rows 0..7 -> acc[0..7] of lane 0; rows 8..15 -> lane 16.
    if (lead) {
        float* dp = dot + (size_t)b * NN + n0 + (lane >> 4) * 8;
        #pragma unroll
        for (int j = 0; j < 8; ++j) dp[j] = acc0[j] + acc1[j];
    }
}

// ---------------------------------------------------------------------------
// Kernel B: per-batch gating pipeline on [N]=2048 (one block per batch).
//   w_c = softmax(beta * dot / (||M_n|| * ||k||))
//   w_g = g*w_c + (1-g)*w_prev
//   w_s = circular 3-tap conv with s
//   w   = w_s^y / sum(w_s^y)
// Also zeroes the r output region (kernel C accumulates into it atomically).
// ---------------------------------------------------------------------------
__device__ __forceinline__ float block_reduce(float v, float* red, int t, int is_max)
{
    red[t] = v;
    __syncthreads();
    #pragma unroll
    for (int sft = 128; sft > 0; sft >>= 1) {
        if (t < sft)
            red[t] = is_max ? fmaxf(red[t], red[t + sft]) : (red[t] + red[t + sft]);
        __syncthreads();
    }
    float r = red[0];
    __syncthreads();
    return r;
}

__global__ __launch_bounds__(256) void ntm_gate_kernel(
    const float* __restrict__ dot, const float* __restrict__ sumsq,
    const float* __restrict__ k,   const float* __restrict__ beta,
    const float* __restrict__ g,   const float* __restrict__ s,
    const float* __restrict__ y,   const float* __restrict__ w_prev,
    float* __restrict__ r_out,     float* __restrict__ w_out)
{
    __shared__ float red[256];
    __shared__ float wsm[NN];

    const int b = blockIdx.x;
    const int t = threadIdx.x;

    // zero r[b,:] for kernel C's atomic accumulation
    r_out[b * DD + t]       = 0.0f;
    r_out[b * DD + t + 256] = 0.0f;

    // ||k||^2
    const float* kb = k + (size_t)b * DD;
    float kv = kb[t] * kb[t] + kb[t + 256] * kb[t + 256];
    const float kn2 = block_reduce(kv, red, t, 0);

    const float bb = beta[b], gb = g[b], yb = y[b];
    const float s0 = s[b * 3 + 0], s1 = s[b * 3 + 1], s2 = s[b * 3 + 2];
    const float* db = dot    + (size_t)b * NN;
    const float* qb = sumsq  + (size_t)b * NN;
    const float* pb = w_prev + (size_t)b * NN;

    float lg[8];
    float mx = -1e30f;
    #pragma unroll
    for (int j = 0; j < 8; ++j) {
        const int i = t + 256 * j;
        const float l = bb * db[i] * rsqrtf(qb[i] * kn2);   // beta * cosine
        lg[j] = l;
        mx = fmaxf(mx, l);
    }
    mx = block_reduce(mx, red, t, 1);

    float se = 0.0f;
    #pragma unroll
    for (int j = 0; j < 8; ++j) { lg[j] = __expf(lg[j] - mx); se += lg[j]; }
    se = block_reduce(se, red, t, 0);
    const float inv = 1.0f / se;

    #pragma unroll
    for (int j = 0; j < 8; ++j) {
        const int i = t + 256 * j;
        wsm[i] = gb * lg[j] * inv + (1.0f - gb) * pb[i];    // interpolate
    }
    __syncthreads();

    float wv[8];
    float sp = 0.0f;
    #pragma unroll
    for (int j = 0; j < 8; ++j) {
        const int i = t + 256 * j;
        float val = s2 * wsm[(i + NN - 1) & (NN - 1)]
                  + s1 * wsm[i]
                  + s0 * wsm[(i + 1) & (NN - 1)];           // circular conv
        val = __powf(val, yb);                              // sharpen
        wv[j] = val;
        sp += val;
    }
    sp = block_reduce(sp, red, t, 0);
    const float invs = 1.0f / sp;

    #pragma unroll
    for (int j = 0; j < 8; ++j) {
        const int i = t + 256 * j;
        w_out[(size_t)b * NN + i] = wv[j] * invs;
    }
}

// ---------------------------------------------------------------------------
// Kernel C: fused read + write, single streaming pass over M.
//   r[b,d]       += sum_n w[b,n] * M[b,n,d]       (atomic f32 reduce across tiles)
//   M_next[b,n,d] = M*(1 - w*e) + w*a             (non-temporal store)
// Grid: (16 row-tiles of 128, BS). Thread t owns d = {2t, 2t+1} -> coalesced float2.
// ---------------------------------------------------------------------------
__global__ __launch_bounds__(256) void ntm_rw_kernel(
    const float* __restrict__ M, const float* __restrict__ e,
    const float* __restrict__ a, const float* __restrict__ w,
    float* __restrict__ r_out,   float* __restrict__ Mn)
{
    __shared__ float wl[128];
    const int b  = blockIdx.y;
    const int t  = threadIdx.x;
    const int n0 = blockIdx.x * 128;

    if (t < 128) wl[t] = w[(size_t)b * NN + n0 + t];
    __syncthreads();

    const int d0 = t * 2;
    const float e0 = e[b * DD + d0], e1 = e[b * DD + d0 + 1];
    const float a0 = a[b * DD + d0], a1 = a[b * DD + d0 + 1];

    const float* Mb = M  + ((size_t)b * NN + n0) * DD + d0;
    float*       Ob = Mn + ((size_t)b * NN + n0) * DD + d0;

    float acc0 = 0.0f, acc1 = 0.0f;
    #pragma unroll 4
    for (int n = 0; n < 128; ++n) {
        __builtin_prefetch(Mb + (size_t)(n + 8) * DD, 0, 1);
        const float wn = wl[n];
        v2f m = __builtin_nontemporal_load((const v2f*)(Mb + (size_t)n * DD));
        acc0 += wn * m.x;
        acc1 += wn * m.y;
        v2f o;
        o.x = m.x * (1.0f - wn * e0) + wn * a0;
        o.y = m.y * (1.0f - wn * e1) + wn * a1;
        __builtin_nontemporal_store(o, (v2f*)(Ob + (size_t)n * DD));
    }
    atomicAdd(&r_out[b * DD + d0],     acc0);
    atomicAdd(&r_out[b * DD + d0 + 1], acc1);
}

// ---------------------------------------------------------------------------
// Host launch
// ---------------------------------------------------------------------------
extern "C" void kernel_launch(void* const* d_in, const int* in_sizes, int n_in,
                              void* d_out, int out_size, void* d_ws, size_t ws_size,
                              hipStream_t stream)
{
    const float* M      = (const float*)d_in[0];
    const float* k      = (const float*)d_in[1];
    const float* beta   = (const float*)d_in[2];
    const float* g      = (const float*)d_in[3];
    const float* s      = (const float*)d_in[4];
    const float* y      = (const float*)d_in[5];
    const float* e      = (const float*)d_in[6];
    const float* a      = (const float*)d_in[7];
    const float* w_prev = (const float*)d_in[8];

    float* out   = (float*)d_out;
    float* r_out = out;                        // [BS, D]
    float* w_out = out + BS * DD;              // [BS, N]
    float* Mn    = out + BS * DD + BS * NN;    // [BS, N, D]

    float* dot   = (float*)d_ws;               // [BS, N]
    float* sumsq = dot + BS * NN;              // [BS, N]

    dim3 gA(NN / 64, BS);
    ntm_content_kernel<<<gA, 128, 0, stream>>>(M, k, dot, sumsq);

    ntm_gate_kernel<<<BS, 256, 0, stream>>>(dot, sumsq, k, beta, g, s, y,
                                            w_prev, r_out, w_out);

    dim3 gC(NN / 128, BS);
    ntm_rw_kernel<<<gC, 256, 0, stream>>>(M, e, a, w_out, r_out, Mn);
}